// TemporalGCN_56873956934012
// MI455X (gfx1250) — compile-verified
//
#include <hip/hip_runtime.h>
#include <hip/hip_bf16.h>

// ---------------------------------------------------------------------------
// TemporalGCN forward for MI455X (gfx1250, wave32)
//   conv1d(9->16,k5,pad2)+relu+pool2 -> conv1d(16->32)+relu+pool2 ->
//   GCN(32->128) -> GCN(128->128) -> mean pool -> linear(128->10)
// GEMMs use V_WMMA_F32_16X16X4_F32 (exact fp32, matrix pipe).
// ---------------------------------------------------------------------------

typedef float v2f __attribute__((ext_vector_type(2)));
typedef float v8f __attribute__((ext_vector_type(8)));

#define NB       64
#define T_IN     4096
#define T_MID    2048      // after pool1
#define T_OUT    1024      // after pool2
#define N_NODES  65536     // 64 * 1024
#define N_EDGES  524288
#define HID      128

// ---------------- conv1 (9->16, k=5, pad=2) + relu + maxpool2 ---------------
__global__ void k_conv1(const float* __restrict__ x, const float* __restrict__ w,
                        const float* __restrict__ bias, float* __restrict__ out) {
    int idx = blockIdx.x * blockDim.x + threadIdx.x;   // 64*16*2048
    if (idx >= NB * 16 * T_MID) return;
    int tp = idx & (T_MID - 1);
    int oc = (idx >> 11) & 15;
    int b  = idx >> 15;
    const float* xb = x + b * 9 * T_IN;
    const float* wo = w + oc * 9 * 5;
    float a0 = bias[oc], a1 = bias[oc];
    int t0 = tp * 2;
    for (int ic = 0; ic < 9; ++ic) {
        const float* xr = xb + ic * T_IN;
        const float* wr = wo + ic * 5;
#pragma unroll
        for (int k = 0; k < 5; ++k) {
            int ta = t0 + k - 2;
            int tb = ta + 1;
            float wv = wr[k];
            float xa = (ta >= 0 && ta < T_IN) ? xr[ta] : 0.0f;
            float xc = (tb >= 0 && tb < T_IN) ? xr[tb] : 0.0f;
            a0 = fmaf(wv, xa, a0);
            a1 = fmaf(wv, xc, a1);
        }
    }
    out[idx] = fmaxf(fmaxf(a0, a1), 0.0f);   // max(relu(a),relu(b))
}

// ---- conv2 (16->32) + relu + maxpool2, written node-major [N_NODES, 32] ----
__global__ void k_conv2(const float* __restrict__ h, const float* __restrict__ w,
                        const float* __restrict__ bias, float* __restrict__ feats) {
    int idx = blockIdx.x * blockDim.x + threadIdx.x;   // 64*32*1024
    if (idx >= NB * 32 * T_OUT) return;
    int tp = idx & (T_OUT - 1);
    int oc = (idx >> 10) & 31;
    int b  = idx >> 15;
    const float* hb = h + b * 16 * T_MID;
    const float* wo = w + oc * 16 * 5;
    float a0 = bias[oc], a1 = bias[oc];
    int t0 = tp * 2;
    for (int ic = 0; ic < 16; ++ic) {
        const float* hr = hb + ic * T_MID;
        const float* wr = wo + ic * 5;
#pragma unroll
        for (int k = 0; k < 5; ++k) {
            int ta = t0 + k - 2;
            int tb = ta + 1;
            float wv = wr[k];
            float xa = (ta >= 0 && ta < T_MID) ? hr[ta] : 0.0f;
            float xc = (tb >= 0 && tb < T_MID) ? hr[tb] : 0.0f;
            a0 = fmaf(wv, xa, a0);
            a1 = fmaf(wv, xc, a1);
        }
    }
    int node = b * T_OUT + tp;
    feats[node * 32 + oc] = fmaxf(fmaxf(a0, a1), 0.0f);
}

// ----------------------------- degree handling ------------------------------
__global__ void k_deg_init(float* __restrict__ deg) {
    int i = blockIdx.x * blockDim.x + threadIdx.x;
    if (i < N_NODES) deg[i] = 1.0f;               // self-loop
}
__global__ void k_deg_count(const int* __restrict__ dst, float* __restrict__ deg) {
    int e = blockIdx.x * blockDim.x + threadIdx.x;
    if (e < N_EDGES) atomicAdd(&deg[dst[e]], 1.0f);
}
__global__ void k_dinv(float* __restrict__ deg) {
    int i = blockIdx.x * blockDim.x + threadIdx.x;
    if (i < N_NODES) deg[i] = rsqrtf(deg[i]);     // in place: deg -> D^{-1/2}
}

// ------------------- GEMM C[M,N] = A[M,K] @ B[K,N] via WMMA -----------------
// One wave computes one 16x16 tile of C. fp32 WMMA 16x16x4.
__global__ void k_gemm_wmma(const float* __restrict__ A, const float* __restrict__ B,
                            float* __restrict__ C, int M, int K, int N) {
    int wave = (blockIdx.x * blockDim.x + threadIdx.x) >> 5;
    int lane = threadIdx.x & 31;
    int ntiles = N >> 4;
    int tm = wave / ntiles;
    int tn = wave - tm * ntiles;
    if (tm * 16 >= M) return;                     // launched exact: never diverges
    int half = lane >> 4;                         // 0: lanes 0-15, 1: lanes 16-31
    int l15  = lane & 15;
    int row  = tm * 16 + l15;                     // A row for this lane
    int col  = tn * 16 + l15;                     // B/C column for this lane
    v8f acc = {};
    for (int k0 = 0; k0 < K; k0 += 4) {
        int ka = k0 + half * 2;
        v2f a, b;
        a[0] = A[row * K + ka];
        a[1] = A[row * K + ka + 1];
        b[0] = B[ka * N + col];
        b[1] = B[(ka + 1) * N + col];
        acc = __builtin_amdgcn_wmma_f32_16x16x4_f32(
            /*neg_a=*/false, a, /*neg_b=*/false, b,
            /*c_mod=*/(short)0, acc, /*reuse_a=*/false, /*reuse_b=*/false);
    }
    int crow = tm * 16 + half * 8;                // C: VGPR r -> M = r + half*8
#pragma unroll
    for (int r = 0; r < 8; ++r)
        C[(crow + r) * N + col] = acc[r];
}

// ------------- aggregation: agg = h * dinv^2 (self-loop) [init] -------------
__global__ void k_agg_init(const float* __restrict__ h, const float* __restrict__ dinv,
                           float* __restrict__ agg) {
    int i = blockIdx.x * blockDim.x + threadIdx.x;   // N_NODES*HID
    if (i >= N_NODES * HID) return;
    float d = dinv[i >> 7];
    agg[i] = h[i] * d * d;
}

// --------- edge scatter: agg[dst] += h[src] * dinv[src]*dinv[dst] -----------
// one wave per edge: 32 lanes x float4 = 128 features
__global__ void k_edge_scatter(const int* __restrict__ src, const int* __restrict__ dst,
                               const float* __restrict__ dinv, const float* __restrict__ h,
                               float* __restrict__ agg) {
    int wave = (blockIdx.x * blockDim.x + threadIdx.x) >> 5;
    int lane = threadIdx.x & 31;
    if (wave >= N_EDGES) return;
    int s = src[wave];
    int d = dst[wave];
    float coef = dinv[s] * dinv[d];
    float4 v = ((const float4*)(h + (size_t)s * HID))[lane];
    float* ap = agg + (size_t)d * HID + lane * 4;
    atomicAdd(ap + 0, v.x * coef);
    atomicAdd(ap + 1, v.y * coef);
    atomicAdd(ap + 2, v.z * coef);
    atomicAdd(ap + 3, v.w * coef);
}

// ------------------------ g = relu(agg + bias), in place --------------------
__global__ void k_finalize(float* __restrict__ agg, const float* __restrict__ bias) {
    int i = blockIdx.x * blockDim.x + threadIdx.x;
    if (i >= N_NODES * HID) return;
    agg[i] = fmaxf(agg[i] + bias[i & (HID - 1)], 0.0f);
}

// ------------------ mean over 1024 nodes + linear(128->10) ------------------
__global__ void k_pool_classify(const float* __restrict__ g, const float* __restrict__ cw,
                                const float* __restrict__ cb, float* __restrict__ out) {
    __shared__ float pooled[HID];
    int b = blockIdx.x;          // 64 graphs
    int f = threadIdx.x;         // 128 features
    const float* gb = g + (size_t)b * T_OUT * HID;
    float s = 0.0f;
    for (int t = 0; t < T_OUT; ++t) s += gb[t * HID + f];
    pooled[f] = s * (1.0f / (float)T_OUT);
    __syncthreads();
    if (f < 10) {
        float acc = cb[f];
#pragma unroll 4
        for (int k = 0; k < HID; ++k) acc = fmaf(pooled[k], cw[k * 10 + f], acc);
        out[b * 10 + f] = acc;
    }
}

// ---------------------------------------------------------------------------
extern "C" void kernel_launch(void* const* d_in, const int* in_sizes, int n_in,
                              void* d_out, int out_size, void* d_ws, size_t ws_size,
                              hipStream_t stream) {
    const float* x   = (const float*)d_in[0];
    const int*   ei  = (const int*)  d_in[1];
    const int*   src = ei;
    const int*   dst = ei + N_EDGES;
    const float* c1w = (const float*)d_in[2];
    const float* c1b = (const float*)d_in[3];
    const float* c2w = (const float*)d_in[4];
    const float* c2b = (const float*)d_in[5];
    const float* g1w = (const float*)d_in[6];
    const float* g1b = (const float*)d_in[7];
    const float* g2w = (const float*)d_in[8];
    const float* g2b = (const float*)d_in[9];
    const float* cw  = (const float*)d_in[10];
    const float* cb  = (const float*)d_in[11];
    float* out = (float*)d_out;

    // workspace layout (floats):
    float* ws    = (float*)d_ws;
    float* h1    = ws;                               // 64*16*2048 = 2,097,152
    float* feats = h1    + (size_t)NB * 16 * T_MID;  // 65536*32   = 2,097,152
    float* bufA  = feats + (size_t)N_NODES * 32;     // 65536*128  = 8,388,608
    float* bufB  = bufA  + (size_t)N_NODES * HID;    // 65536*128  = 8,388,608
    float* deg   = bufB  + (size_t)N_NODES * HID;    // 65536
    (void)ws_size; (void)in_sizes; (void)n_in; (void)out_size;

    const int TPB = 256;

    // Stage 1: temporal convs
    k_conv1<<<(NB * 16 * T_MID) / TPB, TPB, 0, stream>>>(x, c1w, c1b, h1);
    k_conv2<<<(NB * 32 * T_OUT) / TPB, TPB, 0, stream>>>(h1, c2w, c2b, feats);

    // Stage 2: symmetric normalization factors
    k_deg_init <<<N_NODES / TPB, TPB, 0, stream>>>(deg);
    k_deg_count<<<N_EDGES / TPB, TPB, 0, stream>>>(dst, deg);
    k_dinv     <<<N_NODES / TPB, TPB, 0, stream>>>(deg);

    // Stage 3: GCN layer 1 (feats[.,32] @ W1[32,128])
    {
        int waves = (N_NODES / 16) * (HID / 16);     // 32768 waves
        k_gemm_wmma<<<waves * 32 / TPB, TPB, 0, stream>>>(feats, g1w, bufA, N_NODES, 32, HID);
    }
    k_agg_init    <<<(N_NODES * HID) / TPB, TPB, 0, stream>>>(bufA, deg, bufB);
    k_edge_scatter<<<(N_EDGES * 32) / TPB, TPB, 0, stream>>>(src, dst, deg, bufA, bufB);
    k_finalize    <<<(N_NODES * HID) / TPB, TPB, 0, stream>>>(bufB, g1b);   // bufB = g1

    // Stage 4: GCN layer 2 (g1[.,128] @ W2[128,128])
    {
        int waves = (N_NODES / 16) * (HID / 16);
        k_gemm_wmma<<<waves * 32 / TPB, TPB, 0, stream>>>(bufB, g2w, bufA, N_NODES, HID, HID);
    }
    k_agg_init    <<<(N_NODES * HID) / TPB, TPB, 0, stream>>>(bufA, deg, bufB);
    k_edge_scatter<<<(N_EDGES * 32) / TPB, TPB, 0, stream>>>(src, dst, deg, bufA, bufB);
    k_finalize    <<<(N_NODES * HID) / TPB, TPB, 0, stream>>>(bufB, g2b);   // bufB = g2

    // Stage 5: mean pool + classifier
    k_pool_classify<<<NB, HID, 0, stream>>>(bufB, cw, cb, out);
}